// SKConstructor2D_6390911336723
// MI455X (gfx1250) — compile-verified
//
#include <hip/hip_runtime.h>
#include <stdint.h>

// Tree constants derived from the reference setup (complete 8-ary tree).
#define TOPN   585    // (8^4 - 1)/7 : node count of tree levels 0..3 (depth <= 3)
#define LDSN   588    // TOPN padded to a multiple of 4 floats (147 x 16B chunks)
#define NCHUNK 147    // LDSN * 4 bytes / 16 bytes per async b128 chunk
#define MIN_EDGE 0.5f
#define EPSF 1e-9f

__global__ __launch_bounds__(256)
void sk2d_tree_poincare_kernel(const float* __restrict__ offset,
                               const float* __restrict__ cptr,
                               const int*   __restrict__ parent,
                               float2*      __restrict__ out,
                               int L)
{
    // Top-of-tree cache: offsets + parent links for all nodes of depth <= 3.
    __shared__ float s_off[LDSN];
    __shared__ int   s_par[LDSN];

    const int       tid = threadIdx.x;
    const long long gi  = (long long)blockIdx.x * blockDim.x + tid;

    // ---- Stage levels 0..3 into LDS with CDNA5 async global->LDS copies ----
    // (L = 4M >> LDSN, so the 2352-byte reads below are always in bounds.)
    if (tid < NCHUNK) {
        const uint32_t byte    = (uint32_t)tid * 16u;
        // Flat addresses of LDS objects are {aperture_hi32, lds_offset_lo32};
        // the low 32 bits are exactly the wave-relative LDS byte address the
        // async-to-LDS instruction wants in its VDST operand.
        const uint32_t lds_off = ((uint32_t)(uintptr_t)&s_off[0]) + byte;
        const uint32_t lds_par = ((uint32_t)(uintptr_t)&s_par[0]) + byte;
        const uint64_t goff    = (uint64_t)(uintptr_t)offset;
        const uint64_t gpar    = (uint64_t)(uintptr_t)parent;
        // GVS mode: vdst = LDS byte addr, vaddr = 32-bit byte offset, saddr = base.
        asm volatile("global_load_async_to_lds_b128 %0, %1, %2"
                     :: "v"(lds_off), "v"(byte), "s"(goff) : "memory");
        asm volatile("global_load_async_to_lds_b128 %0, %1, %2"
                     :: "v"(lds_par), "v"(byte), "s"(gpar) : "memory");
    }
    asm volatile("s_wait_asynccnt 0" ::: "memory");
    __syncthreads();

    if (gi >= (long long)L) return;
    const int i = (int)gi;

    // ---- Root-path walk: depth <= 8 for L = 4e6, so a direct per-thread
    // ancestor walk beats 22 rounds of pointer-doubling gathers by >2x traffic.
    int   p = i;
    int   d = 0;
    float a = (i < TOPN) ? s_off[i] : offset[i];

    // Deep part of the chain (depths 8..4): global gathers. Sibling groups of
    // 8 share a parent, so these hit the same cachelines; upper levels are
    // L2-resident (whole arrays fit in the 192MB L2).
    while (p >= TOPN) {
        p = parent[p];
        d++;
        a += (p >= TOPN) ? offset[p] : s_off[p];
    }
    // Shallow part (depth <= 3): pure LDS.
    while (p != 0) {
        p = s_par[p];
        d++;
        a += s_off[p];
    }

    // ---- Poincare exp0 along e_x rotated by `a`; ||v|| == r exactly. ----
    const float r   = (float)d * MIN_EDGE;
    const float c   = cptr[0];
    const float sc  = sqrtf(c);
    const float nrm = fmaxf(r, EPSF);
    const float scn = sc * nrm;
    const float fac = tanhf(scn) / scn;
    float sn, cs;
    sincosf(a, &sn, &cs);
    const float fr = fac * r;
    out[i] = make_float2(fr * cs, fr * sn);
}

extern "C" void kernel_launch(void* const* d_in, const int* in_sizes, int n_in,
                              void* d_out, int out_size, void* d_ws, size_t ws_size,
                              hipStream_t stream)
{
    const float* offset = (const float*)d_in[0];
    const float* cptr   = (const float*)d_in[1];
    const int*   parent = (const int*)d_in[2];
    const int    L      = in_sizes[0];

    const int block  = 256;                  // 8 wave32 waves per workgroup
    const int blocks = (L + block - 1) / block;

    sk2d_tree_poincare_kernel<<<blocks, block, 0, stream>>>(
        offset, cptr, parent, (float2*)d_out, L);
}